// GCN_26645977105089
// MI455X (gfx1250) — compile-verified
//
#include <hip/hip_runtime.h>
#include <hip/hip_bf16.h>
#include <math.h>

typedef __attribute__((ext_vector_type(2))) float v2f;
typedef __attribute__((ext_vector_type(8))) float v8f;

#define TPB 256

// ---------- utility ----------
__global__ void zero_f32(float* __restrict__ p, int n) {
  int i = blockIdx.x * blockDim.x + threadIdx.x;
  if (i < n) p[i] = 0.0f;
}

// ---------- degree / normalization ----------
__global__ void deg_kernel(const int* __restrict__ ei, float* __restrict__ deg, int E) {
  int t = blockIdx.x * blockDim.x + threadIdx.x;
  if (t < E) atomicAdd(&deg[ei[E + t]], 1.0f);
}

__global__ void dinv_kernel(const float* __restrict__ deg, float* __restrict__ dinv, int n) {
  int i = blockIdx.x * blockDim.x + threadIdx.x;
  // +1.0f accounts for the self-loop; degree is therefore always >= 1
  if (i < n) dinv[i] = rsqrtf(deg[i] + 1.0f);
}

// ---------- dense transform 1: h1[n,16] = x[n,2] @ W1[2,16] ----------
// One wave handles 16 rows with a single V_WMMA_F32_16X16X4_F32 (K padded 2->4).
// A 16x4 f32 layout: lanes 0-15 hold K=0(.x),1(.y); lanes 16-31 hold K=2,3 (zero pad).
// B 4x16 f32 layout: lanes 0-15 hold rows K=0(.x),1(.y); lanes 16-31 rows K=2,3 (zero).
__global__ void xw1_wmma(const float* __restrict__ x, const float* __restrict__ W1,
                         float* __restrict__ h, int n) {
  int wave = (blockIdx.x * blockDim.x + threadIdx.x) >> 5;
  int lane = threadIdx.x & 31;
  int row0 = wave << 4;
  if (row0 >= n) return;                 // wave-uniform exit: EXEC stays all-1s
  const bool full = (row0 + 16 <= n);    // wave-uniform (always true for n%16==0)
  int m    = lane & 15;
  int half = lane >> 4;
  v2f a, b;
  if (half == 0) {
    int row = row0 + m;
    if (!full && row >= n) row = n - 1;  // clamp only on the (cold) tail path
    a.x = x[row * 2 + 0]; a.y = x[row * 2 + 1];
    b.x = W1[0 * 16 + m]; b.y = W1[1 * 16 + m];
  } else {
    a.x = 0.f; a.y = 0.f; b.x = 0.f; b.y = 0.f;     // K=2,3 zero padding
  }
  v8f c = {};
  c = __builtin_amdgcn_wmma_f32_16x16x4_f32(false, a, false, b, (short)0, c, false, false);
  if (full) {
    // hot path: 8 straight stores, no per-element EXEC churn
    #pragma unroll
    for (int v = 0; v < 8; ++v)
      h[(row0 + v + half * 8) * 16 + m] = c[v];      // D: VGPR v -> M=v (+8 hi half)
  } else {
    #pragma unroll
    for (int v = 0; v < 8; ++v) {
      int r = row0 + v + half * 8;
      if (r < n) h[r * 16 + m] = c[v];
    }
  }
}

// ---------- edge aggregation, 16 features ----------
// thread t = (edge e, feature f): agg[dst*16+f] += h[src*16+f] * dinv[src]*dinv[dst]
// h1/agg1 are 6.4MB each -> resident in 192MB L2; bound by L2 f32-atomic throughput.
// 16 consecutive lanes share an edge: gathers and atomics land as 64B bursts.
__global__ void agg16_kernel(const int* __restrict__ ei, const float* __restrict__ dinv,
                             const float* __restrict__ h, float* __restrict__ agg, int E) {
  int t = blockIdx.x * blockDim.x + threadIdx.x;
  if (t >= E * 16) return;
  int e = t >> 4, f = t & 15;
  int src = ei[e];
  int dst = ei[E + e];
  float norm = dinv[src] * dinv[dst];
  atomicAdd(&agg[dst * 16 + f], h[src * 16 + f] * norm);
}

// ---------- self-loop + bias + ReLU ----------
__global__ void post1_kernel(const float* __restrict__ agg, const float* __restrict__ h,
                             const float* __restrict__ dinv, const float* __restrict__ b1,
                             float* __restrict__ out1, int n) {
  int t = blockIdx.x * blockDim.x + threadIdx.x;
  if (t >= n * 16) return;
  int i = t >> 4, f = t & 15;
  float di = dinv[i];
  float v = agg[t] + di * di * h[t] + b1[f];
  out1[t] = fmaxf(v, 0.0f);
}

// ---------- dense transform 2: h2[n,2] = out1[n,16] @ W2[16,2] ----------
// K=16 via 4 chained V_WMMA_F32_16X16X4_F32; N padded 2->16 with zero B columns.
__global__ void hw2_wmma(const float* __restrict__ out1, const float* __restrict__ W2,
                         float* __restrict__ h2, int n) {
  int wave = (blockIdx.x * blockDim.x + threadIdx.x) >> 5;
  int lane = threadIdx.x & 31;
  int row0 = wave << 4;
  if (row0 >= n) return;                 // wave-uniform
  const bool full = (row0 + 16 <= n);    // wave-uniform
  int m    = lane & 15;
  int half = lane >> 4;
  int row  = row0 + m;
  if (!full && row >= n) row = n - 1;
  v8f c = {};
  #pragma unroll
  for (int k = 0; k < 4; ++k) {
    int kk = 4 * k + half * 2;           // this lane-half supplies K = kk, kk+1
    v2f a, b;
    a.x = out1[row * 16 + kk + 0];
    a.y = out1[row * 16 + kk + 1];
    if (m < 2) { b.x = W2[kk * 2 + m]; b.y = W2[(kk + 1) * 2 + m]; }
    else       { b.x = 0.f;            b.y = 0.f; }               // N pad
    c = __builtin_amdgcn_wmma_f32_16x16x4_f32(false, a, false, b, (short)0, c, false, false);
  }
  if (m < 2) {
    if (full) {
      #pragma unroll
      for (int v = 0; v < 8; ++v)
        h2[(row0 + v + half * 8) * 2 + m] = c[v];
    } else {
      #pragma unroll
      for (int v = 0; v < 8; ++v) {
        int r = row0 + v + half * 8;
        if (r < n) h2[r * 2 + m] = c[v];
      }
    }
  }
}

// ---------- edge aggregation, 2 features ----------
__global__ void agg2_kernel(const int* __restrict__ ei, const float* __restrict__ dinv,
                            const float* __restrict__ h2, float* __restrict__ agg, int E) {
  int t = blockIdx.x * blockDim.x + threadIdx.x;
  if (t >= E * 2) return;
  int e = t >> 1, f = t & 1;
  int src = ei[e];
  int dst = ei[E + e];
  atomicAdd(&agg[dst * 2 + f], h2[src * 2 + f] * dinv[src] * dinv[dst]);
}

// ---------- self-loop + bias + log_softmax (2 classes) ----------
__global__ void final_kernel(const float* __restrict__ agg2, const float* __restrict__ h2,
                             const float* __restrict__ dinv, const float* __restrict__ b2,
                             float* __restrict__ out, int n) {
  int i = blockIdx.x * blockDim.x + threadIdx.x;
  if (i >= n) return;
  float di2 = dinv[i] * dinv[i];
  float o0 = agg2[i * 2 + 0] + di2 * h2[i * 2 + 0] + b2[0];
  float o1 = agg2[i * 2 + 1] + di2 * h2[i * 2 + 1] + b2[1];
  float mx  = fmaxf(o0, o1);
  float lse = mx + logf(expf(o0 - mx) + expf(o1 - mx));
  out[i * 2 + 0] = o0 - lse;
  out[i * 2 + 1] = o1 - lse;
}

extern "C" void kernel_launch(void* const* d_in, const int* in_sizes, int n_in,
                              void* d_out, int out_size, void* d_ws, size_t ws_size,
                              hipStream_t stream) {
  const float* x  = (const float*)d_in[0];
  const int*   ei = (const int*)d_in[1];
  const float* W1 = (const float*)d_in[2];
  const float* b1 = (const float*)d_in[3];
  const float* W2 = (const float*)d_in[4];
  const float* b2 = (const float*)d_in[5];
  float* out = (float*)d_out;

  int n = in_sizes[0] / 2;   // 100000
  int E = in_sizes[1] / 2;   // 3200000

  // workspace carve (floats): deg, dinv, h1, agg1, out1, h2, agg2  (~21.6 MB)
  float* ws   = (float*)d_ws;
  float* deg  = ws;  ws += n;
  float* dinv = ws;  ws += n;
  float* h1   = ws;  ws += (size_t)n * 16;
  float* agg1 = ws;  ws += (size_t)n * 16;
  float* out1 = ws;  ws += (size_t)n * 16;
  float* h2   = ws;  ws += (size_t)n * 2;
  float* agg2 = ws;  ws += (size_t)n * 2;

  auto blocks = [](long long t) { return (unsigned)((t + TPB - 1) / TPB); };

  // zero accumulators every call (graph-replay safe)
  zero_f32<<<blocks(n),                TPB, 0, stream>>>(deg,  n);
  zero_f32<<<blocks((long long)n*16),  TPB, 0, stream>>>(agg1, n * 16);
  zero_f32<<<blocks((long long)n*2),   TPB, 0, stream>>>(agg2, n * 2);

  deg_kernel <<<blocks(E), TPB, 0, stream>>>(ei, deg, E);
  dinv_kernel<<<blocks(n), TPB, 0, stream>>>(deg, dinv, n);

  long long wmma_threads = (long long)((n + 15) / 16) * 32;  // one wave per 16 rows
  xw1_wmma<<<blocks(wmma_threads), TPB, 0, stream>>>(x, W1, h1, n);

  agg16_kernel<<<blocks((long long)E * 16), TPB, 0, stream>>>(ei, dinv, h1, agg1, E);
  post1_kernel<<<blocks((long long)n * 16), TPB, 0, stream>>>(agg1, h1, dinv, b1, out1, n);

  hw2_wmma<<<blocks(wmma_threads), TPB, 0, stream>>>(out1, W2, h2, n);

  agg2_kernel <<<blocks((long long)E * 2), TPB, 0, stream>>>(ei, dinv, h2, agg2, E);
  final_kernel<<<blocks(n), TPB, 0, stream>>>(agg2, h2, dinv, b2, out, n);
}